// TransformerLayer_41205916238264
// MI455X (gfx1250) — compile-verified
//
#include <hip/hip_runtime.h>

// ---------------------------------------------------------------------------
// Bipartite graph attention (GAT-style), MI455X / gfx1250, wave32.
//   Pass 1: items -> users   Pass 2: users -> items
// Projections use fp32 WMMA (V_WMMA_F32_16X16X4_F32) — exact fp32 math,
// GEMM phase is bandwidth-bound at 23.3 TB/s so K=4 throughput is ample.
// Edge phase is gather/atomic bound; q/k/v (<=128MB) live in the 192MB L2.
// ---------------------------------------------------------------------------

constexpr int kNUser = 100000;
constexpr int kNItem = 50000;
constexpr int kNEdge = 400000;
constexpr int kDIn   = 128;   // input dim
constexpr int kDOut  = 128;   // 4 heads * 32 = concat output dim

typedef __attribute__((ext_vector_type(2))) float v2f;
typedef __attribute__((ext_vector_type(8))) float v8f;

// ---------------------------------------------------------------------------
// Projection GEMM: C[n,128] = A[n,128] x B[128,128],
// where B[d][j] = W[h*128*32 + d*32 + e] with j = h*32+e (head-concat cols).
// Block: 256 threads = 8 waves. Each wave owns a 16-row x 128-col strip.
// W staged once into LDS (64 KB) per workgroup.
// ---------------------------------------------------------------------------
__global__ __launch_bounds__(256) void proj_gemm(const float* __restrict__ A,
                                                 const float* __restrict__ W,
                                                 float* __restrict__ C,
                                                 int nrows) {
  __shared__ float Bs[128 * 128];  // Bs[d*128 + j]

  const int tid = threadIdx.x;
  for (int i = tid; i < 128 * 128; i += 256) {
    int d = i >> 7;
    int j = i & 127;
    Bs[i] = W[(j >> 5) * (128 * 32) + d * 32 + (j & 31)];
  }
  __syncthreads();

  const int wave = tid >> 5;
  const int lane = tid & 31;
  const int l16  = lane & 15;
  const int hi   = lane >> 4;          // 0: K pair {0,1} / M rows 0-7; 1: {2,3} / rows 8-15
  const int row0 = blockIdx.x * 128 + wave * 16;

  // A fragment source row (clamped so EXEC stays all-ones; tail masked on store)
  int arow = row0 + l16;
  if (arow >= nrows) arow = nrows - 1;
  const float* ap = A + (size_t)arow * kDIn + 2 * hi;

  v8f acc[8] = {};  // 8 N-tiles of 16 cols, f32 accumulators

#pragma unroll 4
  for (int k = 0; k < 128; k += 4) {
    // A 16x4 frag: lanes 0-15 -> A[m][k],A[m][k+1]; lanes 16-31 -> A[m][k+2],A[m][k+3]
    v2f a = *(const v2f*)(ap + k);
#pragma unroll
    for (int t = 0; t < 8; ++t) {
      // B 4x16 frag: lanes 0-15 -> B[k][n],B[k+1][n]; lanes 16-31 -> B[k+2][n],B[k+3][n]
      int n = t * 16 + l16;
      v2f b;
      b.x = Bs[(k + 2 * hi) * 128 + n];
      b.y = Bs[(k + 2 * hi + 1) * 128 + n];
      acc[t] = __builtin_amdgcn_wmma_f32_16x16x4_f32(
          /*neg_a=*/false, a, /*neg_b=*/false, b,
          /*c_mod=*/(short)0, acc[t], /*reuse_a=*/false, /*reuse_b=*/false);
    }
  }

  // C/D layout: VGPR r -> M = r (lanes 0-15) or M = 8+r (lanes 16-31), N = lane%16
#pragma unroll
  for (int t = 0; t < 8; ++t) {
    int col = t * 16 + l16;
#pragma unroll
    for (int r = 0; r < 8; ++r) {
      int row = row0 + hi * 8 + r;
      if (row < nrows) C[(size_t)row * kDOut + col] = acc[t][r];
    }
  }
}

// ---------------------------------------------------------------------------
// Float atomic max via signed/unsigned integer atomics (init with -inf).
// ---------------------------------------------------------------------------
__device__ inline void atomicMaxFloat(float* addr, float val) {
  if (val >= 0.0f) {
    atomicMax((int*)addr, __float_as_int(val));
  } else {
    atomicMin((unsigned int*)addr, (unsigned int)__float_as_int(val));
  }
}

// Per (edge, head): logit = dot(q[dst,h,:], k[src,h,:]); running max per (dst,h).
__global__ void edge_logit_max(const float* __restrict__ q, const float* __restrict__ kk,
                               const int* __restrict__ dsti, const int* __restrict__ srci,
                               float* __restrict__ logit, float* __restrict__ mbuf,
                               int E, int ndst) {
  int idx = blockIdx.x * blockDim.x + threadIdx.x;
  if (idx >= 4 * E) return;
  int h = idx / E;
  int e = idx - h * E;
  int d = dsti[e], s = srci[e];
  const float4* qp = (const float4*)(q + (size_t)d * kDOut + h * 32);
  const float4* kp = (const float4*)(kk + (size_t)s * kDOut + h * 32);
  float acc = 0.0f;
#pragma unroll
  for (int i = 0; i < 8; ++i) {
    float4 a = qp[i], b = kp[i];
    acc += a.x * b.x + a.y * b.y + a.z * b.z + a.w * b.w;
  }
  logit[idx] = acc;  // idx == h*E + e
  atomicMaxFloat(&mbuf[h * ndst + d], acc);
}

// Per (edge, head): e = exp(logit - max); accumulate per-(dst,h) sums.
__global__ void edge_exp_sum(float* __restrict__ logit, const float* __restrict__ mbuf,
                             float* __restrict__ sbuf, const int* __restrict__ dsti,
                             int E, int ndst) {
  int idx = blockIdx.x * blockDim.x + threadIdx.x;
  if (idx >= 4 * E) return;
  int h = idx / E;
  int e = idx - h * E;
  int d = dsti[e];
  float ex = expf(logit[idx] - mbuf[h * ndst + d]);
  logit[idx] = ex;
  atomicAdd(&sbuf[h * ndst + d], ex);
}

// Per (edge, head, channel): z[dst, h*32+c] += (e/s) * v[src, h*32+c].
__global__ void edge_scatter(const float* __restrict__ att, const float* __restrict__ sbuf,
                             const float* __restrict__ v, const int* __restrict__ dsti,
                             const int* __restrict__ srci, float* __restrict__ z,
                             int E, int ndst) {
  long long idx = blockIdx.x * (long long)blockDim.x + threadIdx.x;
  if (idx >= (long long)E * kDOut) return;
  int e = (int)(idx >> 7);
  int r = (int)(idx & 127);
  int h = r >> 5;
  int d = dsti[e], s = srci[e];
  float a = att[(size_t)h * E + e] / sbuf[h * ndst + d];
  atomicAdd(&z[(size_t)d * kDOut + r], a * v[(size_t)s * kDOut + r]);
}

__global__ void fill_f32(float* __restrict__ p, float val, long long n) {
  long long i = blockIdx.x * (long long)blockDim.x + threadIdx.x;
  if (i < n) p[i] = val;
}

__global__ void relu_f32(float* __restrict__ p, long long n) {
  long long i = blockIdx.x * (long long)blockDim.x + threadIdx.x;
  if (i < n) p[i] = fmaxf(p[i], 0.0f);
}

// ---------------------------------------------------------------------------
extern "C" void kernel_launch(void* const* d_in, const int* in_sizes, int n_in,
                              void* d_out, int out_size, void* d_ws, size_t ws_size,
                              hipStream_t stream) {
  const float* h_user   = (const float*)d_in[0];
  const float* h_item   = (const float*)d_in[1];
  const int*   edge_user = (const int*)d_in[2];
  const int*   edge_item = (const int*)d_in[3];
  const float* u_wq = (const float*)d_in[4];
  const float* u_wk = (const float*)d_in[5];
  const float* u_wv = (const float*)d_in[6];
  const float* i_wq = (const float*)d_in[7];
  const float* i_wk = (const float*)d_in[8];
  const float* i_wv = (const float*)d_in[9];

  float* z_user = (float*)d_out;
  float* z_item = z_user + (size_t)kNUser * kDOut;

  // Workspace carve-out (~163 MB), buffers reused across the two passes.
  float* qd    = (float*)d_ws;                         // [max_n, 128] dst queries
  float* kd    = qd + (size_t)kNUser * kDOut;          // [max_n, 128] src keys
  float* vd    = kd + (size_t)kNUser * kDOut;          // [max_n, 128] src values
  float* logit = vd + (size_t)kNUser * kDOut;          // [4, E] logits -> exp weights
  float* mbuf  = logit + 4 * (size_t)kNEdge;           // [4, max_n] segment max
  float* sbuf  = mbuf + 4 * (size_t)kNUser;            // [4, max_n] segment sum

  const dim3 blk(256);
  auto cdiv = [](long long a, long long b) { return (unsigned int)((a + b - 1) / b); };
  const float neg_inf = -__builtin_inff();

  // Zero the full output accumulator once (harness poisons d_out).
  fill_f32<<<cdiv(out_size, 256), blk, 0, stream>>>((float*)d_out, 0.0f, out_size);

  // ======================= Pass 1: dst = users, src = items ================
  proj_gemm<<<cdiv(kNUser, 128), blk, 0, stream>>>(h_user, u_wq, qd, kNUser);
  proj_gemm<<<cdiv(kNItem, 128), blk, 0, stream>>>(h_item, i_wk, kd, kNItem);
  proj_gemm<<<cdiv(kNItem, 128), blk, 0, stream>>>(h_item, i_wv, vd, kNItem);
  fill_f32<<<cdiv(4LL * kNUser, 256), blk, 0, stream>>>(mbuf, neg_inf, 4LL * kNUser);
  fill_f32<<<cdiv(4LL * kNUser, 256), blk, 0, stream>>>(sbuf, 0.0f, 4LL * kNUser);
  edge_logit_max<<<cdiv(4LL * kNEdge, 256), blk, 0, stream>>>(
      qd, kd, edge_user, edge_item, logit, mbuf, kNEdge, kNUser);
  edge_exp_sum<<<cdiv(4LL * kNEdge, 256), blk, 0, stream>>>(
      logit, mbuf, sbuf, edge_user, kNEdge, kNUser);
  edge_scatter<<<cdiv((long long)kNEdge * kDOut, 256), blk, 0, stream>>>(
      logit, sbuf, vd, edge_user, edge_item, z_user, kNEdge, kNUser);

  // ======================= Pass 2: dst = items, src = users ================
  proj_gemm<<<cdiv(kNItem, 128), blk, 0, stream>>>(h_item, i_wq, qd, kNItem);
  proj_gemm<<<cdiv(kNUser, 128), blk, 0, stream>>>(h_user, u_wk, kd, kNUser);
  proj_gemm<<<cdiv(kNUser, 128), blk, 0, stream>>>(h_user, u_wv, vd, kNUser);
  fill_f32<<<cdiv(4LL * kNItem, 256), blk, 0, stream>>>(mbuf, neg_inf, 4LL * kNItem);
  fill_f32<<<cdiv(4LL * kNItem, 256), blk, 0, stream>>>(sbuf, 0.0f, 4LL * kNItem);
  edge_logit_max<<<cdiv(4LL * kNEdge, 256), blk, 0, stream>>>(
      qd, kd, edge_item, edge_user, logit, mbuf, kNEdge, kNItem);
  edge_exp_sum<<<cdiv(4LL * kNEdge, 256), blk, 0, stream>>>(
      logit, mbuf, sbuf, edge_item, kNEdge, kNItem);
  edge_scatter<<<cdiv((long long)kNEdge * kDOut, 256), blk, 0, stream>>>(
      logit, sbuf, vd, edge_item, edge_user, z_item, kNEdge, kNItem);

  // Final per-head ReLU over both outputs.
  relu_f32<<<cdiv(out_size, 256), blk, 0, stream>>>((float*)d_out, out_size);
}